// MultiHead_60662118088792
// MI455X (gfx1250) — compile-verified
//
#include <hip/hip_runtime.h>

#define B_   8
#define MD_  256
#define N_   4096
#define H_   8
#define F_   16
#define S_   32
#define S3   32768
#define CKV  152   // H*(F+3)
#define WPAD 433   // LDS row stride for conv weights (16 rows x 432, padded)

typedef __attribute__((ext_vector_type(2))) float v2f;
typedef __attribute__((ext_vector_type(8))) float v8f;

__device__ __forceinline__ v8f wmma4(v2f a, v2f b, v8f c) {
  // D = A(16x4 f32) * B(4x16 f32) + C(16x16 f32)
  return __builtin_amdgcn_wmma_f32_16x16x4_f32(false, a, false, b, (short)0, c,
                                               false, false);
}

// ---------------------------------------------------------------------------
// kv = W_kv @ x  : per batch [152x256]*[256x4096]; wave = 16(M) x 64(N) tile
// Invalid A rows (152..159) are address-clamped (never masked): they only
// feed C rows that are never stored, so no exec-predicated loads are needed.
// ---------------------------------------------------------------------------
__global__ void __launch_bounds__(256) gemm_kv_kernel(
    const float* __restrict__ x, const float* __restrict__ W,
    float* __restrict__ kv) {
  const int lane   = threadIdx.x & 31;
  const int waveId = blockIdx.x * 8 + (threadIdx.x >> 5);   // 5120 waves
  const int b   = waveId / 640;
  const int rem = waveId % 640;
  const int mt  = rem / 64;          // 10 M-tiles (152 -> 160 padded)
  const int nt  = rem % 64;          // 64 N-tiles of 64 cols
  const int nbase = nt * 64;
  const int m  = lane & 15;          // A row (M) and B/C column (N) within tile
  const int kk = (lane >> 4) * 2;    // K sub-offset for upper half-lanes
  const int rowA = mt * 16 + m;
  const int rowC = rowA < CKV ? rowA : (CKV - 1);   // clamp: keep loads in-bounds
  const float* wr = W + (size_t)rowC * MD_;
  const float* xb = x + (size_t)b * MD_ * N_;

  v8f c0 = {}, c1 = {}, c2 = {}, c3 = {};
  for (int k0 = 0; k0 < MD_; k0 += 4) {
    const v2f a = *(const v2f*)(wr + k0 + kk);     // 8B-aligned global_load_b64
    const float* xr0 = xb + (size_t)(k0 + kk) * N_;
    const float* xr1 = xr0 + N_;
    const int n0 = nbase + m;
    v2f b0, b1, b2, b3;
    b0.x = xr0[n0];      b0.y = xr1[n0];
    b1.x = xr0[n0 + 16]; b1.y = xr1[n0 + 16];
    b2.x = xr0[n0 + 32]; b2.y = xr1[n0 + 32];
    b3.x = xr0[n0 + 48]; b3.y = xr1[n0 + 48];
    c0 = wmma4(a, b0, c0);
    c1 = wmma4(a, b1, c1);
    c2 = wmma4(a, b2, c2);
    c3 = wmma4(a, b3, c3);
  }
  const int rowHalf = (lane >> 4) * 8;
#pragma unroll
  for (int r = 0; r < 8; ++r) {
    const int row = mt * 16 + r + rowHalf;
    if (row < CKV) {
      float* o = kv + ((size_t)b * CKV + row) * N_ + nbase + m;
      o[0]  = c0[r];
      o[16] = c1[r];
      o[32] = c2[r];
      o[48] = c3[r];
    }
  }
}

// ---------------------------------------------------------------------------
// Training-mode BN statistics over (batch, points): data [B][C][N_]
// ---------------------------------------------------------------------------
__global__ void bn_stats_kernel(const float* __restrict__ data,
                                float* __restrict__ mean,
                                float* __restrict__ rstd, int C) {
  const int c = blockIdx.x;
  float s = 0.f, s2 = 0.f;
  for (int i = threadIdx.x; i < B_ * N_; i += 256) {
    const int b = i >> 12, n = i & (N_ - 1);
    const float v = data[((size_t)b * C + c) * N_ + n];
    s += v; s2 += v * v;
  }
  __shared__ float sh[256], sh2[256];
  sh[threadIdx.x] = s; sh2[threadIdx.x] = s2;
  __syncthreads();
  for (int st = 128; st > 0; st >>= 1) {
    if (threadIdx.x < st) {
      sh[threadIdx.x]  += sh[threadIdx.x + st];
      sh2[threadIdx.x] += sh2[threadIdx.x + st];
    }
    __syncthreads();
  }
  if (threadIdx.x == 0) {
    const float inv = 1.f / (float)(B_ * N_);
    const float mu = sh[0] * inv;
    const float var = sh2[0] * inv - mu * mu;
    mean[c] = mu;
    rstd[c] = rsqrtf(var + 1e-5f);
  }
}

// ---------------------------------------------------------------------------
// Per-point: key BN -> affine -> tanh -> lattice base index + frac
// ---------------------------------------------------------------------------
__global__ void point_kernel(const float* __restrict__ kv,
                             const float* __restrict__ meankv,
                             const float* __restrict__ rstdkv,
                             const float* __restrict__ key_gamma,
                             const float* __restrict__ key_beta,
                             const float* __restrict__ pcd,
                             const float* __restrict__ R,
                             const float* __restrict__ t,
                             int* __restrict__ baseIdx,
                             float* __restrict__ frac) {
  const int tid = blockIdx.x * blockDim.x + threadIdx.x;   // B*H*N
  if (tid >= B_ * H_ * N_) return;
  const int n  = tid & (N_ - 1);
  const int bh = tid >> 12;
  const int h  = bh & 7;
  const int b  = bh >> 3;
  float p[3];
#pragma unroll
  for (int i = 0; i < 3; ++i) {
    const int ch = h * 3 + i;
    const float v = kv[((size_t)b * CKV + ch) * N_ + n];
    const float kr = (v - meankv[ch]) * rstdkv[ch] * key_gamma[ch] + key_beta[ch];
    p[i] = pcd[((size_t)b * 3 + i) * N_ + n] + kr;
  }
  int base = 0;
#pragma unroll
  for (int i = 0; i < 3; ++i) {
    const float key = R[h * 9 + i * 3 + 0] * p[0] + R[h * 9 + i * 3 + 1] * p[1] +
                      R[h * 9 + i * 3 + 2] * p[2] + t[h * 3 + i];
    const float lat = tanhf(key);
    const float pos = (lat + 1.f) * 0.5f * (float)(S_ - 1);
    float flf = floorf(pos);
    flf = fminf(fmaxf(flf, 0.f), (float)(S_ - 2));
    frac[((size_t)bh * 3 + i) * N_ + n] = pos - flf;
    base = base * S_ + (int)flf;
  }
  baseIdx[tid] = base;
}

// ---------------------------------------------------------------------------
// Splat: value BN + trilinear scatter-add onto lattice z[H*F][S^3] (one batch)
// ---------------------------------------------------------------------------
__global__ void splat_kernel(int b, const float* __restrict__ kv,
                             const float* __restrict__ meankv,
                             const float* __restrict__ rstdkv,
                             const float* __restrict__ val_gamma,
                             const float* __restrict__ val_beta,
                             const int* __restrict__ baseIdx,
                             const float* __restrict__ frac,
                             float* __restrict__ z) {
  const int tid = blockIdx.x * blockDim.x + threadIdx.x;   // H*F*N
  if (tid >= H_ * F_ * N_) return;
  const int n  = tid & (N_ - 1);
  const int hf = tid >> 12;
  const int h  = hf >> 4;
  const int ch = H_ * 3 + hf;
  const float v = kv[((size_t)b * CKV + ch) * N_ + n];
  const float val = (v - meankv[ch]) * rstdkv[ch] * val_gamma[hf] + val_beta[hf];
  const int bh = b * H_ + h;
  const int base = baseIdx[bh * N_ + n];
  const float f0 = frac[((size_t)bh * 3 + 0) * N_ + n];
  const float f1 = frac[((size_t)bh * 3 + 1) * N_ + n];
  const float f2 = frac[((size_t)bh * 3 + 2) * N_ + n];
  const float g0 = 1.f - f0, g1 = 1.f - f1, g2 = 1.f - f2;
  float* zp = z + (size_t)hf * S3 + base;
  atomicAdd(zp + 0,                 val * g0 * g1 * g2);
  atomicAdd(zp + 1,                 val * g0 * g1 * f2);
  atomicAdd(zp + S_,                val * g0 * f1 * g2);
  atomicAdd(zp + S_ + 1,            val * g0 * f1 * f2);
  atomicAdd(zp + S_ * S_,           val * f0 * g1 * g2);
  atomicAdd(zp + S_ * S_ + 1,       val * f0 * g1 * f2);
  atomicAdd(zp + S_ * S_ + S_,      val * f0 * f1 * g2);
  atomicAdd(zp + S_ * S_ + S_ + 1,  val * f0 * f1 * f2);
}

// ---------------------------------------------------------------------------
// Head-grouped 3x3x3 conv as implicit GEMM with fp32 WMMA (one batch)
// One block = one head (8 waves, 256 blocks/head); weights staged in LDS.
// Wave = 16 output channels x 16 consecutive z-voxels; K = 27 taps x 16 fi.
// ---------------------------------------------------------------------------
__global__ void __launch_bounds__(256) conv_kernel(
    const float* __restrict__ z, const float* __restrict__ conv_w,
    const float* __restrict__ conv_b, float* __restrict__ zc) {
  const int lane = threadIdx.x & 31;
  const int wv   = threadIdx.x >> 5;
  const int h    = blockIdx.x >> 8;                   // 256 blocks per head
  const int rem  = (blockIdx.x & 255) * 8 + wv;       // 0..2047 voxel tiles
  const int xx   = rem >> 6;
  const int yy   = (rem >> 1) & 31;
  const int zt   = rem & 1;
  const int m    = lane & 15;        // A row (fo) / C column (voxel-in-tile)
  const int kk   = (lane >> 4) * 2;  // K sub-offset for upper half-lanes
  const int z3   = zt * 16 + m;      // this lane's voxel z-coordinate

  // Stage this head's weights [16 fo][432 k] into LDS (padded rows: 433)
  __shared__ float wsh[16 * WPAD];
  const float* wA = conv_w + (size_t)h * 16 * 432;
  for (int i = threadIdx.x; i < 16 * 432; i += 256) {
    const int mm = i / 432, kix = i - mm * 432;
    wsh[mm * WPAD + kix] = wA[i];
  }
  __syncthreads();

  const float* zin = z + (size_t)h * 16 * S3;
  const float* wrow = wsh + m * WPAD;

  v8f acc = {};
  for (int tap = 0; tap < 27; ++tap) {
    const int dx = tap / 9, dy = (tap / 3) % 3, dz = tap % 3;
    const int nx = xx + dx - 1, ny = yy + dy - 1;
    if ((unsigned)nx >= 32u || (unsigned)ny >= 32u) continue;  // wave-uniform
    const int nz  = z3 + dz - 1;
    const bool zok = (unsigned)nz < 32u;
    const int nzc = nz < 0 ? 0 : (nz > 31 ? 31 : nz);          // safe address
    const int nbr = (nx * 32 + ny) * 32 + nzc;
#pragma unroll
    for (int fi0 = 0; fi0 < 16; fi0 += 4) {
      const int fiA = fi0 + kk;
      v2f a, bb;
      a.x = wrow[fiA * 27 + tap];
      a.y = wrow[(fiA + 1) * 27 + tap];
      const float v0 = zin[fiA * S3 + nbr];         // unconditional loads
      const float v1 = zin[(fiA + 1) * S3 + nbr];
      bb.x = zok ? v0 : 0.f;                        // v_cndmask, no exec dance
      bb.y = zok ? v1 : 0.f;
      acc = wmma4(a, bb, acc);
    }
  }
  const int rowHalf = (lane >> 4) * 8;
  const int vox = (xx * 32 + yy) * 32 + zt * 16 + m;
#pragma unroll
  for (int r = 0; r < 8; ++r) {
    const int fo = r + rowHalf;
    zc[((size_t)h * 16 + fo) * S3 + vox] = acc[r] + conv_b[h * 16 + fo];
  }
}

// ---------------------------------------------------------------------------
// Slice: trilinear gather from zc, write pre-BN result to out (one batch)
// ---------------------------------------------------------------------------
__global__ void slice_kernel(int b, const float* __restrict__ zc,
                             const int* __restrict__ baseIdx,
                             const float* __restrict__ frac,
                             float* __restrict__ out) {
  const int tid = blockIdx.x * blockDim.x + threadIdx.x;   // H*F*N
  if (tid >= H_ * F_ * N_) return;
  const int n  = tid & (N_ - 1);
  const int hf = tid >> 12;
  const int h  = hf >> 4;
  const int bh = b * H_ + h;
  const int base = baseIdx[bh * N_ + n];
  const float f0 = frac[((size_t)bh * 3 + 0) * N_ + n];
  const float f1 = frac[((size_t)bh * 3 + 1) * N_ + n];
  const float f2 = frac[((size_t)bh * 3 + 2) * N_ + n];
  const float g0 = 1.f - f0, g1 = 1.f - f1, g2 = 1.f - f2;
  const float* zp = zc + (size_t)hf * S3 + base;
  const float acc =
      zp[0]                * g0 * g1 * g2 + zp[1]                * g0 * g1 * f2 +
      zp[S_]               * g0 * f1 * g2 + zp[S_ + 1]           * g0 * f1 * f2 +
      zp[S_ * S_]          * f0 * g1 * g2 + zp[S_ * S_ + 1]      * f0 * g1 * f2 +
      zp[S_ * S_ + S_]     * f0 * f1 * g2 + zp[S_ * S_ + S_ + 1] * f0 * f1 * f2;
  out[((size_t)b * 128 + hf) * N_ + n] = acc;
}

// ---------------------------------------------------------------------------
// Final BN + ReLU, in place on d_out
// ---------------------------------------------------------------------------
__global__ void bn_relu_kernel(float* __restrict__ out,
                               const float* __restrict__ mean,
                               const float* __restrict__ rstd,
                               const float* __restrict__ gamma,
                               const float* __restrict__ beta) {
  const int tid = blockIdx.x * blockDim.x + threadIdx.x;   // B*128*N
  if (tid >= B_ * 128 * N_) return;
  const int ch = (tid >> 12) & 127;
  float v = out[tid];
  v = (v - mean[ch]) * rstd[ch] * gamma[ch] + beta[ch];
  out[tid] = fmaxf(v, 0.f);
}

extern "C" void kernel_launch(void* const* d_in, const int* in_sizes, int n_in,
                              void* d_out, int out_size, void* d_ws, size_t ws_size,
                              hipStream_t stream) {
  const float* x          = (const float*)d_in[0];
  const float* pcd        = (const float*)d_in[1];
  const float* W_kv       = (const float*)d_in[2];
  const float* key_gamma  = (const float*)d_in[3];
  const float* key_beta   = (const float*)d_in[4];
  const float* val_gamma  = (const float*)d_in[5];
  const float* val_beta   = (const float*)d_in[6];
  const float* R          = (const float*)d_in[7];
  const float* t          = (const float*)d_in[8];
  const float* conv_w     = (const float*)d_in[9];
  const float* conv_b     = (const float*)d_in[10];
  const float* after_gamma= (const float*)d_in[11];
  const float* after_beta = (const float*)d_in[12];
  float* out = (float*)d_out;

  // Workspace carve-up (~58 MB total)
  float* ws      = (float*)d_ws;
  float* kv      = ws;                           // B*CKV*N = 4,980,736 f
  float* meankv  = kv + (size_t)B_ * CKV * N_;   // 256
  float* rstdkv  = meankv + 256;                 // 256
  int*   baseIdx = (int*)(rstdkv + 256);         // B*H*N = 262,144 i32
  float* frac    = (float*)(baseIdx + B_ * H_ * N_); // B*H*3*N = 786,432 f
  float* zbuf    = frac + (size_t)B_ * H_ * 3 * N_;  // H*F*S3 = 4,194,304 f
  float* zcbuf   = zbuf + (size_t)H_ * F_ * S3;      // 4,194,304 f
  float* meanOut = zcbuf + (size_t)H_ * F_ * S3;     // 256
  float* rstdOut = meanOut + 256;                    // 256

  gemm_kv_kernel<<<640, 256, 0, stream>>>(x, W_kv, kv);
  bn_stats_kernel<<<CKV, 256, 0, stream>>>(kv, meankv, rstdkv, CKV);
  point_kernel<<<(B_ * H_ * N_) / 256, 256, 0, stream>>>(
      kv, meankv, rstdkv, key_gamma, key_beta, pcd, R, t, baseIdx, frac);

  for (int b = 0; b < B_; ++b) {
    hipMemsetAsync(zbuf, 0, (size_t)H_ * F_ * S3 * sizeof(float), stream);
    splat_kernel<<<(H_ * F_ * N_) / 256, 256, 0, stream>>>(
        b, kv, meankv, rstdkv, val_gamma, val_beta, baseIdx, frac, zbuf);
    conv_kernel<<<2048, 256, 0, stream>>>(zbuf, conv_w, conv_b, zcbuf);
    slice_kernel<<<(H_ * F_ * N_) / 256, 256, 0, stream>>>(
        b, zcbuf, baseIdx, frac, out);
  }

  bn_stats_kernel<<<128, 256, 0, stream>>>(out, meanOut, rstdOut, 128);
  bn_relu_kernel<<<(B_ * 128 * N_) / 256, 256, 0, stream>>>(
      out, meanOut, rstdOut, after_gamma, after_beta);
}